// FastTextClassifier_65601330479710
// MI455X (gfx1250) — compile-verified
//
#include <hip/hip_runtime.h>
#include <hip/hip_bf16.h>

#define VOCAB 100000
#define EMB_DIM 128
#define HIDDEN 256
#define NUM_CLASSES 20
#define BATCH 4096
#define SEQ 200

typedef float v2f __attribute__((ext_vector_type(2)));
typedef float v8f __attribute__((ext_vector_type(8)));

// ---------------------------------------------------------------------------
// Kernel 1: embedding gather + masked mean pool.
// One wave32 per sample; lane owns 4 consecutive embedding dims (float4).
// ---------------------------------------------------------------------------
__global__ __launch_bounds__(256) void ft_pool_kernel(
    const int* __restrict__ x, const int* __restrict__ lens,
    const float* __restrict__ emb, float* __restrict__ pooled) {
  const int lane = threadIdx.x & 31;
  const int wave = threadIdx.x >> 5;
  const int b = blockIdx.x * 8 + wave;
  if (b >= BATCH) return;

  int len = lens[b];
  if (len < 1) len = 1;
  if (len > SEQ) len = SEQ;

  const int* xb = x + (size_t)b * SEQ;
  float4 acc = make_float4(0.f, 0.f, 0.f, 0.f);

  for (int s0 = 0; s0 < len; s0 += 32) {
    int cnt = len - s0;
    if (cnt > 32) cnt = 32;
    int tok = 0;
    if (lane < cnt) tok = xb[s0 + lane];
    for (int j = 0; j < cnt; ++j) {
      int t = __shfl(tok, j, 32);  // broadcast token j across the wave
      const float4* row = (const float4*)(emb + (size_t)t * EMB_DIM);
      float4 v = row[lane];  // 512B coalesced row read (L2-resident table)
      acc.x += v.x; acc.y += v.y; acc.z += v.z; acc.w += v.w;
    }
  }
  const float inv = 1.0f / (float)len;
  float4 r = make_float4(acc.x * inv, acc.y * inv, acc.z * inv, acc.w * inv);
  ((float4*)(pooled + (size_t)b * EMB_DIM))[lane] = r;
}

// ---------------------------------------------------------------------------
// Kernel 2: hidden = relu(pooled @ W1 + b1) via V_WMMA_F32_16X16X4_F32.
// One wave per 16x16 output tile; K loop = 128/4 = 32 WMMAs.
// ---------------------------------------------------------------------------
__global__ __launch_bounds__(256) void ft_gemm1_kernel(
    const float* __restrict__ pooled, const float* __restrict__ W1,
    const float* __restrict__ b1, float* __restrict__ hidden) {
  const int lane = threadIdx.x & 31;
  const int wave = threadIdx.x >> 5;
  const int tile = blockIdx.x * 8 + wave;      // 4096 tiles total
  const int mt = tile >> 4;                    // 256 row tiles
  const int nt = tile & 15;                    // 16 col tiles
  const int lm = lane & 15;
  const int half = lane >> 4;                  // 0: K,K+1 ; 1: K+2,K+3
  const int kb = half * 2;

  const float* Arow = pooled + (size_t)(mt * 16 + lm) * EMB_DIM;
  const int ncol = nt * 16 + lm;

  v8f c = {0.f, 0.f, 0.f, 0.f, 0.f, 0.f, 0.f, 0.f};
  for (int k0 = 0; k0 < EMB_DIM; k0 += 4) {
    v2f a, bm;
    a.x = Arow[k0 + kb];
    a.y = Arow[k0 + kb + 1];
    bm.x = W1[(size_t)(k0 + kb) * HIDDEN + ncol];
    bm.y = W1[(size_t)(k0 + kb + 1) * HIDDEN + ncol];
    c = __builtin_amdgcn_wmma_f32_16x16x4_f32(
        /*neg_a=*/false, a, /*neg_b=*/false, bm,
        /*c_mod=*/(short)0, c, /*reuse_a=*/false, /*reuse_b=*/false);
  }

  const float bias = b1[ncol];
  const int mbase = mt * 16 + half * 8;
#pragma unroll
  for (int v = 0; v < 8; ++v) {
    float val = c[v] + bias;
    val = val > 0.f ? val : 0.f;  // relu
    hidden[(size_t)(mbase + v) * HIDDEN + ncol] = val;
  }
}

// ---------------------------------------------------------------------------
// Kernel 3: out = hidden @ W2 + b2 (N=20 padded to 32 in-register).
// B-loads are mask-multiplied (branch-free) so EXEC is all-1s at every WMMA.
// ---------------------------------------------------------------------------
__global__ __launch_bounds__(256) void ft_gemm2_kernel(
    const float* __restrict__ hidden, const float* __restrict__ W2,
    const float* __restrict__ b2, float* __restrict__ out) {
  const int lane = threadIdx.x & 31;
  const int wave = threadIdx.x >> 5;
  const int tile = blockIdx.x * 8 + wave;      // 512 tiles total
  const int mt = tile >> 1;                    // 256 row tiles
  const int nt = tile & 1;                     // 2 col tiles (cols 0-15,16-31)
  const int lm = lane & 15;
  const int half = lane >> 4;
  const int kb = half * 2;

  const int ncol = nt * 16 + lm;
  const float nmask = (ncol < NUM_CLASSES) ? 1.0f : 0.0f;  // v_cndmask, no branch
  const int nclamp = (ncol < NUM_CLASSES) ? ncol : (NUM_CLASSES - 1);

  const float* Arow = hidden + (size_t)(mt * 16 + lm) * HIDDEN;

  v8f c = {0.f, 0.f, 0.f, 0.f, 0.f, 0.f, 0.f, 0.f};
  for (int k0 = 0; k0 < HIDDEN; k0 += 4) {
    v2f a, bm;
    a.x = Arow[k0 + kb];
    a.y = Arow[k0 + kb + 1];
    bm.x = W2[(size_t)(k0 + kb) * NUM_CLASSES + nclamp] * nmask;
    bm.y = W2[(size_t)(k0 + kb + 1) * NUM_CLASSES + nclamp] * nmask;
    c = __builtin_amdgcn_wmma_f32_16x16x4_f32(
        false, a, false, bm, (short)0, c, false, false);
  }

  if (ncol < NUM_CLASSES) {  // divergence only after all WMMAs
    const float bias = b2[ncol];
    const int mbase = mt * 16 + half * 8;
#pragma unroll
    for (int v = 0; v < 8; ++v) {
      out[(size_t)(mbase + v) * NUM_CLASSES + ncol] = c[v] + bias;
    }
  }
}

// ---------------------------------------------------------------------------
extern "C" void kernel_launch(void* const* d_in, const int* in_sizes, int n_in,
                              void* d_out, int out_size, void* d_ws, size_t ws_size,
                              hipStream_t stream) {
  const int*   x    = (const int*)d_in[0];     // [B,S] token ids
  const int*   lens = (const int*)d_in[1];     // [B]
  const float* emb  = (const float*)d_in[2];   // [VOCAB,128]
  const float* W1   = (const float*)d_in[3];   // [128,256]
  const float* b1   = (const float*)d_in[4];   // [256]
  const float* W2   = (const float*)d_in[5];   // [256,20]
  const float* b2   = (const float*)d_in[6];   // [20]
  float*       out  = (float*)d_out;           // [B,20]

  float* pooled = (float*)d_ws;                                   // 4096*128*4 = 2 MB
  float* hidden = (float*)((char*)d_ws + (size_t)BATCH * EMB_DIM * sizeof(float)); // 4 MB

  // Kernel 1: 4096 samples, 8 waves/block -> 512 blocks
  ft_pool_kernel<<<BATCH / 8, 256, 0, stream>>>(x, lens, emb, pooled);

  // Kernel 2: (4096/16)*(256/16) = 4096 wave-tiles, 8 waves/block -> 512 blocks
  ft_gemm1_kernel<<<(BATCH / 16) * (HIDDEN / 16) / 8, 256, 0, stream>>>(
      pooled, W1, b1, hidden);

  // Kernel 3: (4096/16)*2 = 512 wave-tiles -> 64 blocks
  ft_gemm2_kernel<<<(BATCH / 16) * 2 / 8, 256, 0, stream>>>(
      hidden, W2, b2, out);
}